// PRMPGATConv_1099511628144
// MI455X (gfx1250) — compile-verified
//
#include <hip/hip_runtime.h>
#include <hip/hip_bf16.h>
#include <math.h>

typedef float v2f __attribute__((ext_vector_type(2)));
typedef float v8f __attribute__((ext_vector_type(8)));

#define SDIM 128
#define HH 4
#define DD 32

// ---------------------------------------------------------------------------
// GEMM: out[m,n] = sum_k A[m,k] * W[n,k]  (+ bias[n]) (optional ReLU)
// K = N = 128 fixed. One wave computes one 16x16 C tile via 32x
// V_WMMA_F32_16X16X4_F32. Block = 256 threads = 8 waves = the 8 N-tiles of a
// 16-row strip. A strip staged into LDS with GLOBAL_LOAD_ASYNC_TO_LDS_B128
// (ASYNCcnt-tracked direct LDS write), W streamed from L2.
// ---------------------------------------------------------------------------
__global__ __launch_bounds__(256) void gemm_nt_128(
    const float* __restrict__ A, const float* __restrict__ W,
    const float* __restrict__ bias, float* __restrict__ out,
    int M, int relu)
{
    __shared__ float As[16 * 128];
    const int tid  = threadIdx.x;
    const int wave = tid >> 5;
    const int lane = tid & 31;
    const int rowBase = blockIdx.x * 16;

    // LDS byte address of As for the async-copy destination operand
    const unsigned as_base =
        (unsigned)(uintptr_t)(__attribute__((address_space(3))) float*)As;

    // stage 16 rows (8KB) of A into LDS: 256 threads x 2 x b128 async copies
    const long maxf = (long)M * SDIM - 4;   // clamp to stay in-bounds (tail rows unused)
    #pragma unroll
    for (int it = 0; it < 2; ++it) {
        const int idx4 = tid + it * 256;            // float4 index within tile
        long fidx = (long)rowBase * SDIM + (long)idx4 * 4;
        if (fidx > maxf) fidx = maxf;
        const float* gptr = A + fidx;
        const unsigned lds_off = as_base + (unsigned)idx4 * 16u;
        asm volatile("global_load_async_to_lds_b128 %0, %1, off"
                     :: "v"(lds_off), "v"(gptr)
                     : "memory");
    }
    asm volatile("s_wait_asynccnt 0x0" ::: "memory");
    __syncthreads();

    const bool hi  = (lane >= 16);
    const int  l16 = lane & 15;
    const int  colTile = wave * 16;
    const int  col = colTile + l16;

    v8f c = {};
    for (int kk = 0; kk < 128; kk += 4) {
        // f32 A-frag 16x4: lanes 0-15 -> K = kk,kk+1 ; lanes 16-31 -> K = kk+2,kk+3
        const int ks = kk + (hi ? 2 : 0);
        v2f a, b;
        a.x = As[l16 * 128 + ks];
        a.y = As[l16 * 128 + ks + 1];
        // B-frag 4x16 (K x N), B[k][n] = W[n*128+k]; same K split across half-waves
        const float* wp = W + (size_t)col * 128 + ks;
        b.x = wp[0];
        b.y = wp[1];
        c = __builtin_amdgcn_wmma_f32_16x16x4_f32(
                /*neg_a=*/false, a, /*neg_b=*/false, b,
                /*c_mod=*/(short)0, c, /*reuse_a=*/false, /*reuse_b=*/false);
    }

    const float bv = bias ? bias[col] : 0.0f;
    #pragma unroll
    for (int v = 0; v < 8; ++v) {
        const int row = rowBase + v + (hi ? 8 : 0);
        if (row < M) {
            float val = c[v] + bv;
            if (relu) val = fmaxf(val, 0.0f);
            out[(size_t)row * 128 + col] = val;
        }
    }
}

// ---------------------------------------------------------------------------
// Fold att_{src,dst} into the projection weights:
//   v[h][k] = sum_d W[(h*32+d)*128 + k] * att[h*32+d]
// ---------------------------------------------------------------------------
__global__ void fold_att_kernel(const float* __restrict__ Wsrc,
                                const float* __restrict__ atts,
                                const float* __restrict__ Wdst,
                                const float* __restrict__ attd,
                                float* __restrict__ vsrc,
                                float* __restrict__ vdst)
{
    const int idx = blockIdx.x * blockDim.x + threadIdx.x;
    if (idx >= 2 * HH * SDIM) return;
    const int which = (idx >= HH * SDIM) ? 1 : 0;
    const int r = idx - which * HH * SDIM;
    const int h = r >> 7, k = r & 127;
    const float* Wm = which ? Wdst : Wsrc;
    const float* am = which ? attd : atts;
    float s = 0.0f;
    for (int d = 0; d < DD; ++d)
        s += Wm[(size_t)(h * DD + d) * SDIM + k] * am[h * DD + d];
    (which ? vdst : vsrc)[h * SDIM + k] = s;
}

// a[n][h] = sum_k X[n][k] * V[h][k]   ([N,128] x [4,128] -> [N,4])
__global__ void node_logits_kernel(const float* __restrict__ X,
                                   const float* __restrict__ V,
                                   float* __restrict__ out, int N)
{
    const int idx = blockIdx.x * blockDim.x + threadIdx.x;
    if (idx >= N * HH) return;
    const int n = idx >> 2, h = idx & 3;
    const float* x = X + (size_t)n * SDIM;
    const float* v = V + h * SDIM;
    float s = 0.0f;
    for (int k = 0; k < SDIM; ++k) s += x[k] * v[k];
    out[idx] = s;
}

__global__ void init_kernel(float* __restrict__ amax,
                            float* __restrict__ denom, int n)
{
    const int i = blockIdx.x * blockDim.x + threadIdx.x;
    if (i < n) { amax[i] = -INFINITY; denom[i] = 0.0f; }
}

// leaky-relu logit per (edge, head), segment-max into amax via float atomicMax
__global__ void edge_logit_max(const float* __restrict__ a_s,
                               const float* __restrict__ a_d,
                               const int* __restrict__ es,
                               const int* __restrict__ ed,
                               float* __restrict__ logit,
                               float* __restrict__ amax, int E)
{
    const int idx = blockIdx.x * blockDim.x + threadIdx.x;
    if (idx >= E * HH) return;
    const int e = idx >> 2, h = idx & 3;
    float a = a_s[es[e] * HH + h] + a_d[ed[e] * HH + h];
    a = (a > 0.0f) ? a : 0.2f * a;          // leaky_relu(0.2)
    logit[idx] = a;
    atomicMax(&amax[ed[e] * HH + h], a);
}

// ex = exp(logit - amax[dst]); segment-sum into denom
__global__ void edge_exp_sum(const int* __restrict__ ed,
                             const float* __restrict__ amax,
                             float* __restrict__ logit_ex,
                             float* __restrict__ denom, int E)
{
    const int idx = blockIdx.x * blockDim.x + threadIdx.x;
    if (idx >= E * HH) return;
    const int e = idx >> 2, h = idx & 3;
    const int d = ed[e] * HH + h;
    const float ex = expf(logit_ex[idx] - amax[d]);
    logit_ex[idx] = ex;
    atomicAdd(&denom[d], ex);
}

// per edge: out[ed] += alpha * xsW[es]   (one wave per edge, each lane owns
// 4 consecutive features -> float4 gathers, all L2-resident).
// The  -predW[ed]*sum(alpha)  term is applied once per node in finalize_kernel
// (sum(alpha) == denom/(denom+1e-16) exactly).
__global__ __launch_bounds__(256) void edge_aggregate(
    const float* __restrict__ xsW,
    const float* __restrict__ ex, const float* __restrict__ denom,
    const int* __restrict__ es, const int* __restrict__ ed,
    float* __restrict__ out, int E)
{
    const int tid = threadIdx.x;
    const int e = blockIdx.x * 8 + (tid >> 5);
    if (e >= E) return;
    const int lane = tid & 31;
    const int s = es[e], d = ed[e];
    const int h = lane >> 3;                 // 8 lanes per head (32 feats)
    const float w = ex[e * HH + h] / (denom[d * HH + h] + 1e-16f);
    const float4 xs = *(const float4*)(xsW + (size_t)s * SDIM + lane * 4);
    float* o = out + (size_t)d * SDIM + lane * 4;
    atomicAdd(o + 0, xs.x * w);
    atomicAdd(o + 1, xs.y * w);
    atomicAdd(o + 2, xs.z * w);
    atomicAdd(o + 3, xs.w * w);
}

// out[d][c] -= predW[d][c] * sum_alpha(d, c/32);  sum_alpha = dn/(dn+1e-16)
__global__ void finalize_kernel(const float* __restrict__ predW,
                                const float* __restrict__ denom,
                                float* __restrict__ out, int Ndst)
{
    const int idx = blockIdx.x * blockDim.x + threadIdx.x;
    if (idx >= Ndst * SDIM) return;
    const int dnode = idx >> 7, c = idx & 127, h = c >> 5;
    const float dn = denom[dnode * HH + h];
    out[idx] -= predW[idx] * (dn / (dn + 1e-16f));
}

extern "C" void kernel_launch(void* const* d_in, const int* in_sizes, int n_in,
                              void* d_out, int out_size, void* d_ws, size_t ws_size,
                              hipStream_t stream)
{
    const float* x_src   = (const float*)d_in[0];
    const float* x_dst   = (const float*)d_in[1];
    const float* W_pred1 = (const float*)d_in[2];
    const float* b_pred1 = (const float*)d_in[3];
    const float* W_pred2 = (const float*)d_in[4];
    const float* b_pred2 = (const float*)d_in[5];
    const float* W_res   = (const float*)d_in[6];
    const float* W_src   = (const float*)d_in[7];
    const float* W_dst   = (const float*)d_in[8];
    const float* att_src = (const float*)d_in[9];
    const float* att_dst = (const float*)d_in[10];
    const float* W_self  = (const float*)d_in[11];
    const float* b_self  = (const float*)d_in[12];
    const int*   edge_src = (const int*)d_in[13];
    const int*   edge_dst = (const int*)d_in[14];

    const int Nsrc = in_sizes[0] / SDIM;
    const int Ndst = in_sizes[1] / SDIM;
    const int E    = in_sizes[13];
    float* out = (float*)d_out;

    // workspace partition (fp32)
    float* ws    = (float*)d_ws;
    float* h1    = ws;  ws += (size_t)Ndst * SDIM;
    float* pred  = ws;  ws += (size_t)Ndst * SDIM;
    float* xsW   = ws;  ws += (size_t)Nsrc * SDIM;
    float* predW = ws;  ws += (size_t)Ndst * SDIM;
    float* vsrc  = ws;  ws += HH * SDIM;
    float* vdst  = ws;  ws += HH * SDIM;
    float* a_s   = ws;  ws += (size_t)Nsrc * HH;
    float* a_d   = ws;  ws += (size_t)Ndst * HH;
    float* amax  = ws;  ws += (size_t)Ndst * HH;
    float* denom = ws;  ws += (size_t)Ndst * HH;
    float* exbuf = ws;  ws += (size_t)E * HH;

    const int gSrc = (Nsrc + 15) / 16;
    const int gDst = (Ndst + 15) / 16;

    init_kernel<<<(Ndst * HH + 255) / 256, 256, 0, stream>>>(amax, denom, Ndst * HH);
    fold_att_kernel<<<(2 * HH * SDIM + 255) / 256, 256, 0, stream>>>(
        W_src, att_src, W_dst, att_dst, vsrc, vdst);

    // node-level WMMA GEMMs
    gemm_nt_128<<<gDst, 256, 0, stream>>>(x_dst, W_pred1, b_pred1, h1, Ndst, 1);
    gemm_nt_128<<<gDst, 256, 0, stream>>>(h1, W_pred2, b_pred2, pred, Ndst, 0);
    gemm_nt_128<<<gSrc, 256, 0, stream>>>(x_src, W_res, nullptr, xsW, Nsrc, 0);
    gemm_nt_128<<<gDst, 256, 0, stream>>>(pred, W_res, nullptr, predW, Ndst, 0);
    gemm_nt_128<<<gDst, 256, 0, stream>>>(x_dst, W_self, b_self, out, Ndst, 0);

    // attention logits per node
    node_logits_kernel<<<(Nsrc * HH + 255) / 256, 256, 0, stream>>>(x_src, vsrc, a_s, Nsrc);
    node_logits_kernel<<<(Ndst * HH + 255) / 256, 256, 0, stream>>>(x_dst, vdst, a_d, Ndst);

    // edge softmax + aggregation
    edge_logit_max<<<(E * HH + 255) / 256, 256, 0, stream>>>(
        a_s, a_d, edge_src, edge_dst, exbuf, amax, E);
    edge_exp_sum<<<(E * HH + 255) / 256, 256, 0, stream>>>(
        edge_dst, amax, exbuf, denom, E);
    edge_aggregate<<<(E + 7) / 8, 256, 0, stream>>>(
        xsW, exbuf, denom, edge_src, edge_dst, out, E);
    finalize_kernel<<<(Ndst * SDIM + 255) / 256, 256, 0, stream>>>(
        predW, denom, out, Ndst);
}